// NodeCorrespondenceSelector_28905129902249
// MI455X (gfx1250) — compile-verified
//
#include <hip/hip_runtime.h>
#include <stdint.h>

#define THRES      0.1f
#define NUM_CORR   256
#define NUM_TGT    2048
#define NUM_SRC    2048
#define CHUNK      4096          // elements per block chunk (16 KB)
#define CPB        1024          // chunks per batch = 2048*2048 / 4096
#define BLK        256           // 8 wave32 per block
#define QPT        4             // b128 (16B) quanta per thread: 4*16B*256thr = 16KB

// ---------------------------------------------------------------------------
// Pass 1: stream the whole map once via CDNA5 async global->LDS copies and
// count thresholded entries per 4096-element chunk. Pure HBM-bandwidth pass.
// ---------------------------------------------------------------------------
__global__ __launch_bounds__(BLK)
void ncs_count_kernel(const float* __restrict__ conf, int* __restrict__ counts)
{
    __shared__ __align__(16) float smem[CHUNK];
    __shared__ int s_total;

    const int tid = threadIdx.x;
    if (tid == 0) s_total = 0;

    const int b     = blockIdx.x / CPB;
    const int chunk = blockIdx.x % CPB;
    const float* base = conf + (size_t)b * ((size_t)NUM_SRC * NUM_TGT)
                             + (size_t)chunk * CHUNK;

    const unsigned ldsBase = (unsigned)(uintptr_t)(&smem[0]);

    // Issue 4 async b128 copies per thread; lanes cover contiguous 16B quanta
    // so each (wave, iter) moves a coalesced 512 B line. Tracked by ASYNCcnt.
#pragma unroll
    for (int k = 0; k < QPT; ++k) {
        const unsigned byteOff = (unsigned)(k * BLK + tid) * 16u;
        const unsigned ldsAddr = ldsBase + byteOff;
        asm volatile("global_load_async_to_lds_b128 %0, %1, %2"
                     :
                     : "v"(ldsAddr), "v"(byteOff), "s"(base)
                     : "memory");
    }
    // Wait for this wave's own async copies before reading its quanta back.
    asm volatile("s_wait_asynccnt 0" ::: "memory");
    __syncthreads();   // also publishes s_total = 0

    int c = 0;
#pragma unroll
    for (int k = 0; k < QPT; ++k) {
        const unsigned e = (unsigned)(k * BLK + tid) * 4u;   // float index
        const float4 v = *reinterpret_cast<const float4*>(&smem[e]);
        c += (v.x <= THRES) + (v.y <= THRES) + (v.z <= THRES) + (v.w <= THRES);
    }
    atomicAdd(&s_total, c);
    __syncthreads();
    if (tid == 0) counts[blockIdx.x] = s_total;
}

// ---------------------------------------------------------------------------
// Pass 2: per batch, exclusive prefix-sum over the 1024 chunk counts.
// Also zero-initializes the output slice (harness poisons d_out).
// ---------------------------------------------------------------------------
__global__ __launch_bounds__(1024)
void ncs_scan_kernel(const int* __restrict__ counts, int* __restrict__ bases,
                     int* __restrict__ out)
{
    __shared__ int s[CPB];
    const int b = blockIdx.x;
    const int t = threadIdx.x;

    const int c = counts[b * CPB + t];
    s[t] = c;
    __syncthreads();
#pragma unroll
    for (int stride = 1; stride < CPB; stride <<= 1) {
        const int add = (t >= stride) ? s[t - stride] : 0;
        __syncthreads();
        s[t] += add;
        __syncthreads();
    }
    bases[b * CPB + t] = s[t] - c;   // exclusive prefix = starting rank

    if (t < NUM_CORR * 2) out[b * NUM_CORR * 2 + t] = 0;
}

// ---------------------------------------------------------------------------
// Pass 3: chunks whose starting rank < 256 (typically one per batch) re-read
// their data and emit (src, tgt) index pairs in ascending flat-index order,
// matching jax.lax.top_k's lowest-index-first tie-break on the zeroed values.
// ---------------------------------------------------------------------------
__global__ __launch_bounds__(BLK)
void ncs_emit_kernel(const float* __restrict__ conf, const int* __restrict__ bases,
                     int* __restrict__ out)
{
    const int base = bases[blockIdx.x];
    if (base >= NUM_CORR) return;    // uniform per block: safe early exit

    __shared__ int s[BLK];
    const int tid   = threadIdx.x;
    const int b     = blockIdx.x / CPB;
    const int chunk = blockIdx.x % CPB;

    // Thread t owns 16 consecutive elements -> block scan preserves index order.
    const float4* p = reinterpret_cast<const float4*>(
        conf + (size_t)b * ((size_t)NUM_SRC * NUM_TGT)
             + (size_t)chunk * CHUNK + (size_t)tid * 16);

    float v[16];
#pragma unroll
    for (int k = 0; k < 4; ++k) {
        const float4 q = p[k];
        v[4 * k + 0] = q.x; v[4 * k + 1] = q.y;
        v[4 * k + 2] = q.z; v[4 * k + 3] = q.w;
    }

    bool m[16];
    int c = 0;
#pragma unroll
    for (int j = 0; j < 16; ++j) { m[j] = (v[j] <= THRES); c += m[j]; }

    s[tid] = c;
    __syncthreads();
#pragma unroll
    for (int stride = 1; stride < BLK; stride <<= 1) {
        const int add = (tid >= stride) ? s[tid - stride] : 0;
        __syncthreads();
        s[tid] += add;
        __syncthreads();
    }

    int r = base + (s[tid] - c);                 // this thread's first rank
    const int flatBase = chunk * CHUNK + tid * 16;
#pragma unroll
    for (int j = 0; j < 16; ++j) {
        if (m[j]) {
            if (r < NUM_CORR) {
                const int flat = flatBase + j;
                out[(b * NUM_CORR + r) * 2 + 0] = flat >> 11;            // / 2048
                out[(b * NUM_CORR + r) * 2 + 1] = flat & (NUM_TGT - 1);  // % 2048
            }
            ++r;
        }
    }
}

// ---------------------------------------------------------------------------
extern "C" void kernel_launch(void* const* d_in, const int* in_sizes, int n_in,
                              void* d_out, int out_size, void* d_ws, size_t ws_size,
                              hipStream_t stream)
{
    const float* conf = (const float*)d_in[0];
    int* out = (int*)d_out;

    const int batch = in_sizes[0] / (NUM_SRC * NUM_TGT);   // 8
    const int nchunks = batch * CPB;                       // 8192

    int* counts = (int*)d_ws;           // nchunks ints
    int* bases  = counts + nchunks;     // nchunks ints  (64 KB total)

    ncs_count_kernel<<<nchunks, BLK, 0, stream>>>(conf, counts);
    ncs_scan_kernel <<<batch, CPB, 0, stream>>>(counts, bases, out);
    ncs_emit_kernel <<<nchunks, BLK, 0, stream>>>(conf, bases, out);
}